// HadamardTransform_8031588844225
// MI455X (gfx1250) — compile-verified
//
#include <hip/hip_runtime.h>
#include <hip/hip_bf16.h>

typedef __attribute__((ext_vector_type(2))) float v2f;
typedef __attribute__((ext_vector_type(8))) float v8f;

// Block-wise Hadamard transform: y = x.reshape(-1,128) @ H, exact fp32 via
// V_WMMA_F32_16X16X4_F32.
//
// Each wave computes a 16-row x 128-col output tile. Loop nest is K-outer:
// all 8 N-tile accumulators stay in registers (64 VGPRs), and each K-step
// loads one A fragment (global, 2 VGPRs) + 8 B fragments (LDS, 16 VGPRs)
// and issues 8 interleaved WMMAs. Dependent WMMAs on the same accumulator
// are 8 instructions apart, hiding the WMMA pipeline latency, and register
// pressure stays ~110 VGPRs -> no spills at 6 waves/SIMD.
//
// WMMA fragment layouts (CDNA5 ISA 7.12.2, 32-bit data):
//   A (16x4):  VGPR0: lanes 0-15 -> (M=lane, K=0), lanes 16-31 -> (M=lane-16, K=2)
//              VGPR1: K=1 / K=3.   => lane loads 2 consecutive floats of its row.
//   B (4x16):  VGPR0: lanes 0-15 -> (K=0, N=lane), lanes 16-31 -> (K=2, N=lane-16)
//              VGPR1: K=1 / K=3.
//   C/D (16x16): VGPR v: lanes 0-15 -> (M=v, N=lane), lanes 16-31 -> (M=v+8, N=lane-16)

__global__ __launch_bounds__(256)
__attribute__((amdgpu_waves_per_eu(6))) void
hadamard_wmma_f32(const float* __restrict__ x,
                  const float* __restrict__ H,
                  float* __restrict__ y,
                  int num_row_blocks) {
  // H pre-swizzled into WMMA B-fragment order, k-major so the 8 fragments of
  // one K-step are 256B apart per lane (ds_load_2addr_b64-combinable):
  // sB[((k*8 + n)*32 + lane)*2 + c] =
  //     H[(4k + c + (lane>=16 ? 2 : 0))*128 + n*16 + (lane&15)]
  __shared__ float sB[32 * 8 * 32 * 2]; // 64 KB (of 320 KB per WGP)

  const int tid  = threadIdx.x;
  const int lane = tid & 31;
  const int wave = tid >> 5;

  for (int idx = tid; idx < 16384; idx += 256) {
    int c = idx & 1;
    int l = (idx >> 1) & 31;
    int n = (idx >> 6) & 7;
    int k = idx >> 9;
    int row = 4 * k + c + ((l >= 16) ? 2 : 0);
    int col = n * 16 + (l & 15);
    sB[idx] = H[row * 128 + col];
  }
  __syncthreads();

  const int gwave  = blockIdx.x * 8 + wave;
  const int nwaves = gridDim.x * 8;
  const int mrow   = lane & 15;          // M row within tile for this lane
  const int khalf  = (lane >> 4) * 2;    // lanes 16-31 take K+2 elements

  // Per-lane base into the swizzled B image.
  const float* sBlane = &sB[lane * 2];

  for (int rb = gwave; rb < num_row_blocks; rb += nwaves) {
    const float* xrow = x + (size_t)rb * 2048 + (size_t)mrow * 128 + khalf;

    v8f acc[8];
#pragma unroll
    for (int n = 0; n < 8; ++n) acc[n] = (v8f){};

#pragma unroll 4
    for (int k = 0; k < 32; ++k) {
      // One A fragment per K-step; streamed (never reused) -> non-temporal.
      v2f a = __builtin_nontemporal_load((const v2f*)(xrow + 4 * k));
#pragma unroll
      for (int n = 0; n < 8; ++n) {
        v2f b = *(const v2f*)(sBlane + (k * 8 + n) * 64); // ds_load_b64, 256B apart
        acc[n] = __builtin_amdgcn_wmma_f32_16x16x4_f32(
            /*neg_a=*/false, a,
            /*neg_b=*/false, b,
            /*c_mod=*/(short)0, acc[n],
            /*reuse_a=*/false, /*reuse_b=*/false);
      }
    }

    float* yrow = y + (size_t)rb * 2048;
    const int drow = ((lane >= 16) ? 8 : 0);
#pragma unroll
    for (int n = 0; n < 8; ++n) {
#pragma unroll
      for (int v = 0; v < 8; ++v) {
        __builtin_nontemporal_store(
            acc[n][v], &yrow[(size_t)(v + drow) * 128 + n * 16 + mrow]);
      }
    }
  }
}

extern "C" void kernel_launch(void* const* d_in, const int* in_sizes, int n_in,
                              void* d_out, int out_size, void* d_ws, size_t ws_size,
                              hipStream_t stream) {
  const float* x = (const float*)d_in[0];  // 8192*8192 fp32
  const float* H = (const float*)d_in[1];  // 128*128 fp32
  float* y = (float*)d_out;

  const long total = (long)in_sizes[0];
  const int num_row_blocks = (int)(total / 2048);  // 16 rows x 128 cols each

  // 1024 blocks x 8 waves = 8192 waves; 32768 row blocks -> 4 per wave,
  // amortizing the per-block LDS swizzle of H.
  int grid = 1024;
  if (num_row_blocks < 8 * grid) grid = (num_row_blocks + 7) / 8;
  if (grid < 1) grid = 1;

  hadamard_wmma_f32<<<grid, 256, 0, stream>>>(x, H, y, num_row_blocks);
}